// FocalLoss_59416577572895
// MI455X (gfx1250) — compile-verified
//
#include <hip/hip_runtime.h>

typedef float v2f __attribute__((ext_vector_type(2)));
typedef float v8f __attribute__((ext_vector_type(8)));

#define BB 16
#define CC 8
#define MM 32
#define FEPS 1e-4f

__global__ __launch_bounds__(256)
void retina_main(const float* __restrict__ cls,
                 const float* __restrict__ regs,
                 const float* __restrict__ anchors,
                 const float* __restrict__ ann,
                 float* __restrict__ pf, float* __restrict__ pr, float* __restrict__ pn,
                 int A, int nPart)
{
    __shared__ float s_ann[MM * 5];
    __shared__ float s_a2[MM];
    __shared__ float s_wf[8], s_wr[8], s_wn[8];

    const int tid = threadIdx.x;
    const int img = blockIdx.y;

    if (tid < MM * 5) s_ann[tid] = ann[img * MM * 5 + tid];
    __syncthreads();
    if (tid < MM) {
        float x1 = s_ann[tid * 5 + 0], y1 = s_ann[tid * 5 + 1];
        float x2 = s_ann[tid * 5 + 2], y2 = s_ann[tid * 5 + 3];
        s_a2[tid] = (x2 - x1 + 1.0f) * (y2 - y1 + 1.0f);
    }
    __syncthreads();

    const int a   = blockIdx.x * 256 + tid;
    const bool act = (a < A);
    const int aa  = act ? a : 0;

    // anchor box for IoU comes from anchors[img] (reference: anch = anchors[i])
    const float4 ab = ((const float4*)anchors)[(size_t)img * A + aa];
    const float a1 = (ab.z - ab.x + 1.0f) * (ab.w - ab.y + 1.0f);

    float best = -1.0f;
    int   bidx = 0;
    #pragma unroll 4
    for (int m = 0; m < MM; ++m) {
        float bx1 = s_ann[m * 5 + 0], by1 = s_ann[m * 5 + 1];
        float bx2 = s_ann[m * 5 + 2], by2 = s_ann[m * 5 + 3];
        float lab = s_ann[m * 5 + 4];
        float w = fminf(ab.z, bx2) - fmaxf(ab.x, bx1) + 1.0f;
        float h = fminf(ab.w, by2) - fmaxf(ab.y, by1) + 1.0f;
        w = fmaxf(w, 0.0f);
        h = fmaxf(h, 0.0f);
        float inter = w * h;
        float iou = inter / (a1 + s_a2[m] - inter);
        iou = (lab > -0.5f) ? iou : -1.0f;        // mask invalid annotations
        if (iou > best) { best = iou; bidx = m; } // first-occurrence argmax
    }

    const bool pos = act && (best >= 0.4f);
    const bool neg = act && (best < 0.3f);

    // ---------------- focal classification loss ----------------
    float facc = 0.0f;
    if (pos || neg) {
        const float4* c4 = (const float4*)(cls + ((size_t)img * A + aa) * CC);
        float4 c0 = c4[0], c1 = c4[1];
        float pv[CC] = {c0.x, c0.y, c0.z, c0.w, c1.x, c1.y, c1.z, c1.w};
        int lab = pos ? (int)s_ann[bidx * 5 + 4] : -1;
        #pragma unroll
        for (int c = 0; c < CC; ++c) {
            float p = fminf(fmaxf(pv[c], FEPS), 1.0f - FEPS);
            bool isone = (c == lab);
            float q  = isone ? (1.0f - p) : p;
            float fw = (isone ? 0.25f : 0.75f) * q * q;          // gamma = 2
            float bce = isone ? (-logf(p)) : (-log1pf(-p));
            facc += fw * bce;
        }
    }

    // ---------------- smooth-L1 regression loss ----------------
    float racc = 0.0f;
    if (pos) {
        const float4 rg = ((const float4*)regs)[(size_t)img * A + aa];
        const float4 a0 = ((const float4*)anchors)[aa];  // geometry from anchors[0]
        float aw = a0.z - a0.x, ah = a0.w - a0.y;
        float acx = a0.x + 0.5f * aw, acy = a0.y + 0.5f * ah;
        float gx1 = s_ann[bidx * 5 + 0], gy1 = s_ann[bidx * 5 + 1];
        float gx2 = s_ann[bidx * 5 + 2], gy2 = s_ann[bidx * 5 + 3];
        float w0 = gx2 - gx1, h0 = gy2 - gy1;
        float gcx = gx1 + 0.5f * w0, gcy = gy1 + 0.5f * h0;  // unclamped for centers
        float gw = fmaxf(w0, 1.0f), gh = fmaxf(h0, 1.0f);    // clamped for logs
        float t0 = ((gcx - acx) / aw) / 0.1f;
        float t1 = ((gcy - acy) / ah) / 0.1f;
        float t2 = logf(gw / aw) / 0.2f;
        float t3 = logf(gh / ah) / 0.2f;
        float d0 = fabsf(t0 - rg.x), d1 = fabsf(t1 - rg.y);
        float d2 = fabsf(t2 - rg.z), d3 = fabsf(t3 - rg.w);
        const float th = 1.0f / 9.0f, sub = 0.5f / 9.0f;
        racc  = (d0 <= th) ? 4.5f * d0 * d0 : d0 - sub;
        racc += (d1 <= th) ? 4.5f * d1 * d1 : d1 - sub;
        racc += (d2 <= th) ? 4.5f * d2 * d2 : d2 - sub;
        racc += (d3 <= th) ? 4.5f * d3 * d3 : d3 - sub;
    }
    const float nacc = pos ? 1.0f : 0.0f;

    // ---- wave-level reduction via V_WMMA_F32_16X16X4_F32 (ones × data) ----
    // A = all-ones 16x4 => D[m][n] = colsum(n). Each lane duplicates its
    // scalar into both B components, so Sum_{all 32 lanes} c[0] = 4 * wave total
    // (layout-robust: lane L's two B values land in column L%16 in any layout).
    v2f ones; ones.x = 1.0f; ones.y = 1.0f;
    v2f bf; bf.x = facc; bf.y = facc;
    v2f br; br.x = racc; br.y = racc;
    v2f bn; bn.x = nacc; bn.y = nacc;
    v8f cf = {}, cr = {}, cn = {};
    cf = __builtin_amdgcn_wmma_f32_16x16x4_f32(false, ones, false, bf, (short)0, cf, false, false);
    cr = __builtin_amdgcn_wmma_f32_16x16x4_f32(false, ones, false, br, (short)0, cr, false, false);
    cn = __builtin_amdgcn_wmma_f32_16x16x4_f32(false, ones, false, bn, (short)0, cn, false, false);

    float wf = cf[0], wr = cr[0], wn = cn[0];
    #pragma unroll
    for (int off = 16; off > 0; off >>= 1) {
        wf += __shfl_xor(wf, off, 32);
        wr += __shfl_xor(wr, off, 32);
        wn += __shfl_xor(wn, off, 32);
    }

    if ((tid & 31) == 0) {
        int wv = tid >> 5;
        s_wf[wv] = wf * 0.25f;
        s_wr[wv] = wr * 0.25f;
        s_wn[wv] = wn * 0.25f;
    }
    __syncthreads();
    if (tid == 0) {
        float F = 0.0f, R = 0.0f, N = 0.0f;
        #pragma unroll
        for (int w = 0; w < 8; ++w) { F += s_wf[w]; R += s_wr[w]; N += s_wn[w]; }
        int slot = img * nPart + blockIdx.x;
        pf[slot] = F;
        pr[slot] = R;
        pn[slot] = N;
    }
}

__global__ __launch_bounds__(256)
void retina_finalize(const float* __restrict__ pf, const float* __restrict__ pr,
                     const float* __restrict__ pn, const float* __restrict__ ann,
                     float* __restrict__ out, int nPart)
{
    __shared__ float sf[256], sr[256], sn[256];
    const int tid = threadIdx.x;
    float clsTot = 0.0f, regTot = 0.0f;

    for (int img = 0; img < BB; ++img) {
        float f = 0.0f, r = 0.0f, n = 0.0f;
        for (int j = tid; j < nPart; j += 256) {
            f += pf[img * nPart + j];
            r += pr[img * nPart + j];
            n += pn[img * nPart + j];
        }
        __syncthreads();
        sf[tid] = f; sr[tid] = r; sn[tid] = n;
        __syncthreads();
        for (int s = 128; s > 0; s >>= 1) {
            if (tid < s) {
                sf[tid] += sf[tid + s];
                sr[tid] += sr[tid + s];
                sn[tid] += sn[tid + s];
            }
            __syncthreads();
        }
        if (tid == 0) {
            int nv = 0;
            for (int m = 0; m < MM; ++m)
                nv += (ann[img * MM * 5 + m * 5 + 4] > -0.5f) ? 1 : 0;
            float npos = sn[0];
            float cl = (nv > 0) ? sf[0] / fmaxf(npos, 1.0f) : 0.0f;
            float rl = (nv > 0 && npos > 0.0f) ? sr[0] / (npos * 4.0f) : 0.0f;
            clsTot += cl;
            regTot += rl;
        }
    }
    if (tid == 0) {
        out[0] = clsTot * (1.0f / BB);
        out[1] = regTot * (1.0f / BB);
    }
}

extern "C" void kernel_launch(void* const* d_in, const int* in_sizes, int n_in,
                              void* d_out, int out_size, void* d_ws, size_t ws_size,
                              hipStream_t stream)
{
    (void)n_in; (void)out_size; (void)ws_size;
    const float* cls     = (const float*)d_in[0];  // [B,A,C]
    const float* regs    = (const float*)d_in[1];  // [B,A,4]
    const float* anchors = (const float*)d_in[2];  // [B,A,4]
    const float* ann     = (const float*)d_in[3];  // [B,M,5]
    float* out = (float*)d_out;                    // [2]

    const int A = in_sizes[0] / (BB * CC);
    const int nPart = (A + 255) / 256;

    float* pf = (float*)d_ws;
    float* pr = pf + (size_t)BB * nPart;
    float* pn = pr + (size_t)BB * nPart;

    retina_main<<<dim3(nPart, BB), 256, 0, stream>>>(cls, regs, anchors, ann,
                                                     pf, pr, pn, A, nPart);
    retina_finalize<<<1, 256, 0, stream>>>(pf, pr, pn, ann, out, nPart);
}